// PointNetObject_4956392259994
// MI455X (gfx1250) — compile-verified
//
#include <hip/hip_runtime.h>

typedef __attribute__((ext_vector_type(2))) float v2f;
typedef __attribute__((ext_vector_type(8))) float v8f;

// ---------------------------------------------------------------------------
// Elementwise transposes
// ---------------------------------------------------------------------------

// (B,C,S) -> (B,S,C), indexed over output
__global__ void k_tr_bcs_to_bsc(const float* __restrict__ in, float* __restrict__ out,
                                int B, int C, int S) {
  long i = (long)blockIdx.x * blockDim.x + threadIdx.x;
  long total = (long)B * C * S;
  if (i >= total) return;
  int c = (int)(i % C);
  long r = i / C;
  int s = (int)(r % S);
  int b = (int)(r / S);
  out[i] = in[((long)b * C + c) * S + s];
}

// (B,S,C) -> (B,C,S), indexed over output
__global__ void k_tr_bsc_to_bcs(const float* __restrict__ in, float* __restrict__ out,
                                int B, int S, int C) {
  long i = (long)blockIdx.x * blockDim.x + threadIdx.x;
  long total = (long)B * S * C;
  if (i >= total) return;
  int s = (int)(i % S);
  long r = i / S;
  int c = (int)(r % C);
  int b = (int)(r / C);
  out[i] = in[((long)b * S + s) * C + c];
}

__global__ void k_fill_zero(float* __restrict__ p, int n) {
  int i = blockIdx.x * blockDim.x + threadIdx.x;
  if (i < n) p[i] = 0.0f;
}

// ---------------------------------------------------------------------------
// Farthest point sampling: one block per batch. dist[] lives in LDS.
// Matches lax.scan semantics: idx[0]=0, idx[i]=argmax of running min-dist.
// ---------------------------------------------------------------------------
__global__ void k_fps(const float* __restrict__ xyz /*(B,n,3)*/, int n, int npoint,
                      int* __restrict__ fidx /*(B,npoint)*/) {
  extern __shared__ char smraw[];
  float* dist = (float*)smraw;                    // n floats
  float* rmax = dist + n;                         // blockDim floats
  int*   ridx = (int*)(rmax + blockDim.x);        // blockDim ints
  __shared__ int s_far;

  int b = blockIdx.x, t = threadIdx.x, T = blockDim.x;
  const float* base = xyz + (long)b * n * 3;

  for (int i = t; i < n; i += T) dist[i] = 1.0e10f;
  if (t == 0) s_far = 0;
  __syncthreads();

  for (int it = 0; it < npoint; ++it) {
    int far = s_far;
    if (t == 0) fidx[(long)b * npoint + it] = far;
    float cx = base[(long)far * 3 + 0];
    float cy = base[(long)far * 3 + 1];
    float cz = base[(long)far * 3 + 2];
    float bestv = -1.0f;
    int besti = 0;
    for (int i = t; i < n; i += T) {
      float dx = base[(long)i * 3 + 0] - cx;
      float dy = base[(long)i * 3 + 1] - cy;
      float dz = base[(long)i * 3 + 2] - cz;
      float d = dx * dx + dy * dy + dz * dz;
      float nd = fminf(dist[i], d);
      dist[i] = nd;
      if (nd > bestv) { bestv = nd; besti = i; }   // strided -> lowest idx kept
    }
    rmax[t] = bestv; ridx[t] = besti;
    __syncthreads();
    for (int s = T >> 1; s > 0; s >>= 1) {
      if (t < s) {
        float ov = rmax[t + s]; int oi = ridx[t + s];
        if (ov > rmax[t] || (ov == rmax[t] && oi < ridx[t])) { rmax[t] = ov; ridx[t] = oi; }
      }
      __syncthreads();
    }
    if (t == 0) s_far = ridx[0];
    __syncthreads();
  }
}

// new_xyz[b][s][c] = xyz[b][fidx[b][s]][c]
__global__ void k_gather_newxyz(const float* __restrict__ xyz, const int* __restrict__ fidx,
                                float* __restrict__ nxyz, int B, int n, int np) {
  long i = (long)blockIdx.x * blockDim.x + threadIdx.x;
  long total = (long)B * np * 3;
  if (i >= total) return;
  int c = (int)(i % 3);
  long r = i / 3;
  int s = (int)(r % np);
  int b = (int)(r / np);
  int g = fidx[(long)b * np + s];
  nxyz[i] = xyz[((long)b * n + g) * 3 + c];
}

// ---------------------------------------------------------------------------
// Ball query: one wave32 per (b,s). Appends in-radius indices in ascending
// order using ballot+popcount prefix; fills remainder with first index.
// Uses the reference's expanded |a|^2+|b|^2-2ab distance.
// ---------------------------------------------------------------------------
__global__ void k_ball_query(const float* __restrict__ xyz /*(B,n,3)*/,
                             const float* __restrict__ nxyz /*(B,np,3)*/,
                             int* __restrict__ gi /*(B,np,ns)*/,
                             int n, int np, int ns, float r2) {
  int lane = threadIdx.x;           // 0..31
  int s = blockIdx.x, b = blockIdx.y;
  const float* cb = nxyz + ((long)b * np + s) * 3;
  float cx = cb[0], cy = cb[1], cz = cb[2];
  float cn = cx * cx + cy * cy + cz * cz;
  int* out = gi + ((long)b * np + s) * ns;
  const float* pb = xyz + (long)b * n * 3;

  int cnt = 0;
  int firstIdx = -1;
  for (int basei = 0; basei < n; basei += 32) {
    int i = basei + lane;
    float px = pb[(long)i * 3 + 0];
    float py = pb[(long)i * 3 + 1];
    float pz = pb[(long)i * 3 + 2];
    float pn = px * px + py * py + pz * pz;
    float sqr = cn + pn - 2.0f * (cx * px + cy * py + cz * pz);
    bool pred = !(sqr > r2);
    unsigned mask = (unsigned)__ballot(pred);
    if (firstIdx < 0 && mask) firstIdx = basei + __ffs(mask) - 1;
    int rank = __popc(mask & ((1u << lane) - 1u));
    if (pred && (cnt + rank) < ns) out[cnt + rank] = i;
    cnt += __popc(mask);
    if (cnt >= ns) break;
  }
  int m = cnt < ns ? cnt : ns;
  if (firstIdx < 0) firstIdx = 0;   // cannot happen (centroid in-radius), safety
  for (int j = m + lane; j < ns; j += 32) out[j] = firstIdx;
}

// ---------------------------------------------------------------------------
// Build grouped features, K-padded to Cp:
//   out[b][s][k][0:3]    = xyz[gi]-new_xyz
//   out[b][s][k][3:Cin]  = feat[gi]
//   out[b][s][k][Cin:Cp] = 0
// ---------------------------------------------------------------------------
__global__ void k_group_concat(const float* __restrict__ xyz /*(B,n,3)*/,
                               const float* __restrict__ feat /*(B,n,Cf)*/,
                               const float* __restrict__ nxyz /*(B,np,3)*/,
                               const int* __restrict__ gi /*(B,np,ns)*/,
                               float* __restrict__ out /*(B,np,ns,Cp)*/,
                               int B, int n, int np, int ns, int Cf, int Cp) {
  int Cin = 3 + Cf;
  long total = (long)B * np * ns * Cp;
  long i = (long)blockIdx.x * blockDim.x + threadIdx.x;
  if (i >= total) return;
  int c = (int)(i % Cp);
  long r = i / Cp;
  int k = (int)(r % ns);
  long r2 = r / ns;
  int s = (int)(r2 % np);
  int b = (int)(r2 / np);
  float v = 0.0f;
  if (c < Cin) {
    int g = gi[((long)b * np + s) * ns + k];
    if (c < 3) v = xyz[((long)b * n + g) * 3 + c] - nxyz[((long)b * np + s) * 3 + c];
    else       v = feat[((long)b * n + g) * Cf + (c - 3)];
  }
  out[i] = v;
}

// SA3 concat, K-padded to Cp: out[b][k][0:3]=xyz, [3:Cin]=feat, [Cin:Cp]=0
__global__ void k_concat_all(const float* __restrict__ xyz /*(B,n,3)*/,
                             const float* __restrict__ feat /*(B,n,Cf)*/,
                             float* __restrict__ out, int B, int n, int Cf, int Cp) {
  int Cin = 3 + Cf;
  long total = (long)B * n * Cp;
  long i = (long)blockIdx.x * blockDim.x + threadIdx.x;
  if (i >= total) return;
  int c = (int)(i % Cp);
  long r = i / Cp;
  int k = (int)(r % n);
  int b = (int)(r / n);
  float v = 0.0f;
  if (c < Cin) {
    v = (c < 3) ? xyz[((long)b * n + k) * 3 + c]
                : feat[((long)b * n + k) * Cf + (c - 3)];
  }
  out[i] = v;
}

// Transpose + zero-pad weights: Wt[n][k] = (k<K) ? W[k][n] : 0, Wt is N x Kp
__global__ void k_prep_w(const float* __restrict__ W, float* __restrict__ Wt,
                         int K, int Kp, int N) {
  long i = (long)blockIdx.x * blockDim.x + threadIdx.x;
  long total = (long)N * Kp;
  if (i >= total) return;
  int k = (int)(i % Kp);
  int nn = (int)(i / Kp);
  Wt[i] = (k < K) ? W[(long)k * N + nn] : 0.0f;
}

// ---------------------------------------------------------------------------
// GEMM: out[M,N] = A[M,Kp] @ Wt[N,Kp]^T + bias[N], V_WMMA_F32_16X16X4_F32.
// Kp % 4 == 0 (zero-padded), so the inner loop is branch-free and both the
// A pair and B pair are contiguous 8B-aligned -> global_load_b64 each.
// Block = 32x4: four waves cover four M-tiles of one N-tile.
// Layouts per CDNA5 ISA 7.12.2 (32-bit A 16x4, 32-bit C/D 16x16).
// ---------------------------------------------------------------------------
__global__ void k_gemm_bias_wmma(const float* __restrict__ A, const float* __restrict__ Wt,
                                 const float* __restrict__ bias, float* __restrict__ out,
                                 int M, int Kp, int N) {
  int lane = threadIdx.x;                      // 0..31
  int tn = blockIdx.x;
  int tm = blockIdx.y * 4 + threadIdx.y;
  int half = (lane < 16) ? 0 : 2;              // K sub-offset for this lane half
  int row = tm * 16 + (lane & 15);             // A row this lane feeds
  int col = tn * 16 + (lane & 15);             // B/C/D column this lane holds

  const float* ap = A + (long)row * Kp + half;   // 8B aligned (Kp%4==0, half even)
  const float* bp = Wt + (long)col * Kp + half;

  v8f acc = {0.f, 0.f, 0.f, 0.f, 0.f, 0.f, 0.f, 0.f};
  for (int kb = 0; kb < Kp; kb += 4) {
    v2f a = *(const v2f*)(ap + kb);
    v2f bm = *(const v2f*)(bp + kb);
    acc = __builtin_amdgcn_wmma_f32_16x16x4_f32(false, a, false, bm,
                                                (short)0, acc, false, false);
  }

  float bc = bias[col];
#pragma unroll
  for (int v = 0; v < 8; ++v) {
    int r = tm * 16 + ((lane < 16) ? v : v + 8);
    out[(long)r * N + col] = acc[v] + bc;
  }
}

// ---------------------------------------------------------------------------
// Per-column mean/var over all M rows (matches jnp.mean/var over axes 0,1,2)
// ---------------------------------------------------------------------------
__global__ void k_colstats(const float* __restrict__ x, long M, int C,
                           float* __restrict__ mean, float* __restrict__ var) {
  __shared__ float ssum[256], ssq[256];
  int c = blockIdx.x, t = threadIdx.x;
  float s = 0.0f, q = 0.0f;
  for (long r = t; r < M; r += 256) {
    float v = x[r * C + c];
    s += v; q += v * v;
  }
  ssum[t] = s; ssq[t] = q;
  __syncthreads();
  for (int st = 128; st > 0; st >>= 1) {
    if (t < st) { ssum[t] += ssum[t + st]; ssq[t] += ssq[t + st]; }
    __syncthreads();
  }
  if (t == 0) {
    float m = ssum[0] / (float)M;
    mean[c] = m;
    var[c] = ssq[0] / (float)M - m * m;
  }
}

__global__ void k_bn_relu(float* __restrict__ x, long total, int C,
                          const float* __restrict__ mean, const float* __restrict__ var,
                          const float* __restrict__ g, const float* __restrict__ be) {
  long i = (long)blockIdx.x * blockDim.x + threadIdx.x;
  if (i >= total) return;
  int c = (int)(i % C);
  float v = g[c] * (x[i] - mean[c]) * rsqrtf(var[c] + 1e-5f) + be[c];
  x[i] = fmaxf(v, 0.0f);
}

// max over the sample axis: (BS, ns, C) -> (BS, C)
__global__ void k_maxpool(const float* __restrict__ x, float* __restrict__ out,
                          int BS, int ns, int C) {
  long i = (long)blockIdx.x * blockDim.x + threadIdx.x;
  long total = (long)BS * C;
  if (i >= total) return;
  int c = (int)(i % C);
  int bs = (int)(i / C);
  const float* p = x + (long)bs * ns * C + c;
  float m = p[0];
  for (int k = 1; k < ns; ++k) m = fmaxf(m, p[(long)k * C]);
  out[i] = m;
}

// ---------------------------------------------------------------------------
// Host orchestration
// ---------------------------------------------------------------------------
static inline unsigned cdiv(long a, long b) { return (unsigned)((a + b - 1) / b); }

// A has row stride Kp (Kp%4==0, zero-padded); logical K <= Kp.
static void mlp_layer(const float* A, float* out, long M, int K, int Kp, int N,
                      const float* w, const float* bias, const float* g, const float* be,
                      float* wtbuf, float* mean, float* var, hipStream_t stream) {
  k_prep_w<<<cdiv((long)N * Kp, 256), 256, 0, stream>>>(w, wtbuf, K, Kp, N);
  dim3 grid((unsigned)(N / 16), (unsigned)(M / 64));
  k_gemm_bias_wmma<<<grid, dim3(32, 4), 0, stream>>>(A, wtbuf, bias, out, (int)M, Kp, N);
  k_colstats<<<N, 256, 0, stream>>>(out, M, N, mean, var);
  long total = M * (long)N;
  k_bn_relu<<<cdiv(total, 256), 256, 0, stream>>>(out, total, N, mean, var, g, be);
}

extern "C" void kernel_launch(void* const* d_in, const int* in_sizes, int n_in,
                              void* d_out, int out_size, void* d_ws, size_t ws_size,
                              hipStream_t stream) {
  const int B = 16, N = 4096, HC = 64;

  const float* l0_xyz = (const float*)d_in[0];
  const float* l0_pts = (const float*)d_in[1];
  const float* W[6]  = { (const float*)d_in[2],  (const float*)d_in[6],
                         (const float*)d_in[10], (const float*)d_in[14],
                         (const float*)d_in[18], (const float*)d_in[22] };
  const float* Bi[6] = { (const float*)d_in[3],  (const float*)d_in[7],
                         (const float*)d_in[11], (const float*)d_in[15],
                         (const float*)d_in[19], (const float*)d_in[23] };
  const float* G[6]  = { (const float*)d_in[4],  (const float*)d_in[8],
                         (const float*)d_in[12], (const float*)d_in[16],
                         (const float*)d_in[20], (const float*)d_in[24] };
  const float* Be[6] = { (const float*)d_in[5],  (const float*)d_in[9],
                         (const float*)d_in[13], (const float*)d_in[17],
                         (const float*)d_in[21], (const float*)d_in[25] };

  float* ws = (float*)d_ws;
  // workspace layout (floats)
  float* xyzt0 = ws + 0;            // 16*4096*3 = 196608
  float* ptst0 = ws + 196608;       // 196608
  float* nx1   = ws + 393216;       // 16*256*3  = 12288
  float* nx2   = ws + 405504;       // 16*128*3  = 6144
  float* feat1 = ws + 411648;       // 16*256*128 = 524288
  float* feat2 = ws + 935936;       // 16*128*256 = 524288
  float* meanb = ws + 1460224;      // 512
  float* varb  = ws + 1460736;      // 512
  int*   fidx  = (int*)(ws + 1461248);  // up to 4096 ints
  int*   gi    = (int*)(ws + 1465344);  // up to 131072 ints
  float* wtbuf = ws + 1596416;      // 131072 (max N*Kp = 512*256)
  float* bufB  = ws + 1727488;      // 16777216 (GEMM-1 outputs)
  float* bufS  = ws + 18504704;     // 33554432 shared: grouped feats / GEMM-2 out
  (void)ws_size; (void)n_in; (void)in_sizes; (void)out_size;

  float* outp = (float*)d_out;
  float* o_l1xyz = outp;            // 16*3*256 = 12288
  float* o_l1pts = outp + 12288;    // 16*128*256 = 524288
  float* o_l2xyz = outp + 536576;   // 16*3*128 = 6144
  float* o_l2pts = outp + 542720;   // 16*256*128 = 524288
  float* o_l3xyz = outp + 1067008;  // 48
  float* o_x     = outp + 1067056;  // 16*512 = 8192

  // ---- input transposes: (B,3,N) -> (B,N,3) ----
  {
    long t = (long)B * 3 * N;
    k_tr_bcs_to_bsc<<<cdiv(t, 256), 256, 0, stream>>>(l0_xyz, xyzt0, B, 3, N);
    k_tr_bcs_to_bsc<<<cdiv(t, 256), 256, 0, stream>>>(l0_pts, ptst0, B, 3, N);
  }

  // ================= SA1: n=4096 -> np=256, ns=32, 6->64->128 =================
  {
    int n = N, np = 256, ns = 32, Cf = 3, Cin = 6, Cp = 8, C1 = HC, C2 = HC * 2;
    size_t shm = (size_t)(n + 2 * 256) * 4;
    k_fps<<<B, 256, shm, stream>>>(xyzt0, n, np, fidx);
    k_gather_newxyz<<<cdiv((long)B * np * 3, 256), 256, 0, stream>>>(xyzt0, fidx, nx1, B, n, np);
    k_ball_query<<<dim3(np, B), 32, 0, stream>>>(xyzt0, nx1, gi, n, np, ns, 0.2f * 0.2f);
    long M = (long)B * np * ns;                     // 131072
    k_group_concat<<<cdiv(M * Cp, 256), 256, 0, stream>>>(xyzt0, ptst0, nx1, gi, bufS,
                                                          B, n, np, ns, Cf, Cp);
    mlp_layer(bufS, bufB, M, Cin, Cp, C1, W[0], Bi[0], G[0], Be[0], wtbuf, meanb, varb, stream);
    mlp_layer(bufB, bufS, M, C1, C1, C2, W[1], Bi[1], G[1], Be[1], wtbuf, meanb, varb, stream);
    k_maxpool<<<cdiv((long)B * np * C2, 256), 256, 0, stream>>>(bufS, feat1, B * np, ns, C2);
    // outputs
    k_tr_bsc_to_bcs<<<cdiv((long)B * np * 3, 256), 256, 0, stream>>>(nx1, o_l1xyz, B, np, 3);
    k_tr_bsc_to_bcs<<<cdiv((long)B * np * C2, 256), 256, 0, stream>>>(feat1, o_l1pts, B, np, C2);
  }

  // ================= SA2: n=256 -> np=128, ns=64, 131->128->256 ===============
  {
    int n = 256, np = 128, ns = 64, Cf = 128, Cin = 131, Cp = 132, C1 = HC * 2, C2 = HC * 4;
    size_t shm = (size_t)(n + 2 * 256) * 4;
    k_fps<<<B, 256, shm, stream>>>(nx1, n, np, fidx);
    k_gather_newxyz<<<cdiv((long)B * np * 3, 256), 256, 0, stream>>>(nx1, fidx, nx2, B, n, np);
    k_ball_query<<<dim3(np, B), 32, 0, stream>>>(nx1, nx2, gi, n, np, ns, 0.25f * 0.25f);
    long M = (long)B * np * ns;                     // 131072
    k_group_concat<<<cdiv(M * Cp, 256), 256, 0, stream>>>(nx1, feat1, nx2, gi, bufS,
                                                          B, n, np, ns, Cf, Cp);
    mlp_layer(bufS, bufB, M, Cin, Cp, C1, W[2], Bi[2], G[2], Be[2], wtbuf, meanb, varb, stream);
    mlp_layer(bufB, bufS, M, C1, C1, C2, W[3], Bi[3], G[3], Be[3], wtbuf, meanb, varb, stream);
    k_maxpool<<<cdiv((long)B * np * C2, 256), 256, 0, stream>>>(bufS, feat2, B * np, ns, C2);
    k_tr_bsc_to_bcs<<<cdiv((long)B * np * 3, 256), 256, 0, stream>>>(nx2, o_l2xyz, B, np, 3);
    k_tr_bsc_to_bcs<<<cdiv((long)B * np * C2, 256), 256, 0, stream>>>(feat2, o_l2pts, B, np, C2);
  }

  // ================= SA3 (group_all): n=128, 259->256->512 ====================
  {
    int n = 128, Cf = 256, Cin = 259, Cp = 260, C1 = HC * 4, C2 = HC * 8;
    long M = (long)B * n;                           // 2048
    k_concat_all<<<cdiv(M * Cp, 256), 256, 0, stream>>>(nx2, feat2, bufS, B, n, Cf, Cp);
    mlp_layer(bufS, bufB, M, Cin, Cp, C1, W[4], Bi[4], G[4], Be[4], wtbuf, meanb, varb, stream);
    mlp_layer(bufB, bufS, M, C1, C1, C2, W[5], Bi[5], G[5], Be[5], wtbuf, meanb, varb, stream);
    k_maxpool<<<cdiv((long)B * C2, 256), 256, 0, stream>>>(bufS, o_x, B, n, C2);
    k_fill_zero<<<1, 64, 0, stream>>>(o_l3xyz, 48);
  }
}